// MixedN_25958782337848
// MI455X (gfx1250) — compile-verified
//
#include <hip/hip_runtime.h>
#include <hip/hip_bf16.h>

typedef __attribute__((ext_vector_type(2))) float v2f;
typedef __attribute__((ext_vector_type(8))) float v8f;

#define NNODES 100000
#define NEDGES 1600000
#define NFEAT  128
#define NOUT   64
#define NGRAPH 64

// ---------------------------------------------------------------- utilities
__global__ void zero4_kernel(float4* __restrict__ p, long long n4) {
    long long i = (long long)blockIdx.x * blockDim.x + threadIdx.x;
    if (i < n4) p[i] = make_float4(0.f, 0.f, 0.f, 0.f);
}

__global__ void deg_kernel(const int* __restrict__ dst, float* __restrict__ deg, int E) {
    int e = blockIdx.x * blockDim.x + threadIdx.x;
    if (e < E) unsafeAtomicAdd(&deg[dst[e]], 1.0f);
}

// dinv = rsqrt(deg_from_edges + 1)  (the +1 is the self loop; always > 0)
__global__ void dinv_kernel(float* __restrict__ deg, int n) {
    int i = blockIdx.x * blockDim.x + threadIdx.x;
    if (i < n) deg[i] = rsqrtf(deg[i] + 1.0f);
}

// ---------------------------------------------------------------- WMMA GEMM
// C[M,NCOLS] = act(A[M,128] * B[128,NCOLS] + bias), fp32 WMMA 16x16x4.
// Block = (NCOLS/16) waves; each wave owns a 16-column stripe and 4 row tiles
// (64 rows per block). A tile staged in LDS with stride 132 (bank-conflict free).
// Full-tile blocks take an unguarded store path (only the last block is ragged).
template <int NCOLS, bool BIAS, bool RELU>
__global__ __launch_bounds__((NCOLS / 16) * 32)
void gemm_wmma(const float* __restrict__ A, const float* __restrict__ B,
               const float* __restrict__ bias, float* __restrict__ C, int M) {
    constexpr int K = 128;
    constexpr int LDA = K + 4;              // floats; 132 mod 64 banks = 4
    __shared__ float lds[64 * LDA];

    const int tid = threadIdx.x;
    const int rowBase = blockIdx.x * 64;

    // Cooperative load of a 64 x 128 A tile (fully coalesced float4s).
    for (int i = tid * 4; i < 64 * K; i += blockDim.x * 4) {
        int r = i >> 7;                     // /128
        int c = i & (K - 1);
        float4 v = make_float4(0.f, 0.f, 0.f, 0.f);
        int gr = rowBase + r;
        if (gr < M) v = *(const float4*)(A + (long long)gr * K + c);
        *(float4*)&lds[r * LDA + c] = v;
    }
    __syncthreads();

    const int wave  = tid >> 5;
    const int lane  = tid & 31;
    const int lsub  = lane & 15;
    const int khalf = lane >> 4;            // 0: K,K+1   1: K+2,K+3
    const int col   = wave * 16 + lsub;

    v8f acc[4] = {};                         // 4 row tiles x 16x16 f32

#pragma unroll 4
    for (int k0 = 0; k0 < K; k0 += 4) {
        const int kr = k0 + khalf * 2;
        v2f b;
        b.x = B[kr * NCOLS + col];
        b.y = B[(kr + 1) * NCOLS + col];
        v2f a0 = *(const v2f*)&lds[(0 * 16 + lsub) * LDA + kr];
        v2f a1 = *(const v2f*)&lds[(1 * 16 + lsub) * LDA + kr];
        v2f a2 = *(const v2f*)&lds[(2 * 16 + lsub) * LDA + kr];
        v2f a3 = *(const v2f*)&lds[(3 * 16 + lsub) * LDA + kr];
        acc[0] = __builtin_amdgcn_wmma_f32_16x16x4_f32(false, a0, false, b, (short)0, acc[0], false, false);
        acc[1] = __builtin_amdgcn_wmma_f32_16x16x4_f32(false, a1, false, b, (short)0, acc[1], false, false);
        acc[2] = __builtin_amdgcn_wmma_f32_16x16x4_f32(false, a2, false, b, (short)0, acc[2], false, false);
        acc[3] = __builtin_amdgcn_wmma_f32_16x16x4_f32(false, a3, false, b, (short)0, acc[3], false, false);
    }

    float bval = 0.f;
    if (BIAS) bval = bias[col];

    // D layout: VGPR j -> row = base + khalf*8 + j, col = lane&15.
    float* __restrict__ cptr = C + (long long)(rowBase + khalf * 8) * NCOLS + col;

    if (rowBase + 64 <= M) {
        // Fast path: whole 64-row tile in range -> unguarded stores with
        // immediate offsets from a single base (no per-element EXEC dance).
#pragma unroll
        for (int r = 0; r < 4; ++r) {
#pragma unroll
            for (int j = 0; j < 8; ++j) {
                float v = acc[r][j] + bval;
                if (RELU) v = fmaxf(v, 0.f);
                cptr[(long long)(r * 16 + j) * NCOLS] = v;
            }
        }
    } else {
#pragma unroll
        for (int r = 0; r < 4; ++r) {
#pragma unroll
            for (int j = 0; j < 8; ++j) {
                int grow = rowBase + r * 16 + khalf * 8 + j;
                if (grow < M) {
                    float v = acc[r][j] + bval;
                    if (RELU) v = fmaxf(v, 0.f);
                    C[(long long)grow * NCOLS + col] = v;
                }
            }
        }
    }
}

// ---------------------------------------------------------------- edge scatter
// agg[dst] += t[src] * dinv[src]*dinv[dst]   (32 threads per edge, float4 each)
__global__ void scatter_kernel(const float* __restrict__ t, const int* __restrict__ src,
                               const int* __restrict__ dst, const float* __restrict__ dinv,
                               float* __restrict__ agg) {
    long long idx = (long long)blockIdx.x * blockDim.x + threadIdx.x;
    if (idx >= (long long)NEDGES * 32) return;
    int e = (int)(idx >> 5);
    int c = (int)(idx & 31) << 2;
    int s = src[e], d = dst[e];
    float w = dinv[s] * dinv[d];
    float4 v = *(const float4*)(t + (long long)s * NFEAT + c);
    float* o = agg + (long long)d * NFEAT + c;
    unsafeAtomicAdd(o + 0, v.x * w);
    unsafeAtomicAdd(o + 1, v.y * w);
    unsafeAtomicAdd(o + 2, v.z * w);
    unsafeAtomicAdd(o + 3, v.w * w);
}

// agg = relu(agg + t * dinv^2 + bias)   (fused self-loop + bias + relu)
__global__ void combine_kernel(float* __restrict__ agg, const float* __restrict__ t,
                               const float* __restrict__ dinv, const float* __restrict__ bias) {
    long long idx = (long long)blockIdx.x * blockDim.x + threadIdx.x;
    if (idx >= (long long)NNODES * 32) return;
    int n = (int)(idx >> 5);
    int c = (int)(idx & 31) << 2;
    float di = dinv[n];
    float w = di * di;
    float4 a = *(const float4*)(agg + (long long)n * NFEAT + c);
    float4 h = *(const float4*)(t + (long long)n * NFEAT + c);
    float4 b = *(const float4*)(bias + c);
    float4 o;
    o.x = fmaxf(a.x + h.x * w + b.x, 0.f);
    o.y = fmaxf(a.y + h.y * w + b.y, 0.f);
    o.z = fmaxf(a.z + h.z * w + b.z, 0.f);
    o.w = fmaxf(a.w + h.w * w + b.w, 0.f);
    *(float4*)(agg + (long long)n * NFEAT + c) = o;
}

// ---------------------------------------------------------------- pooling
__global__ void pool_count_kernel(const int* __restrict__ batch, float* __restrict__ counts) {
    int n = blockIdx.x * blockDim.x + threadIdx.x;
    if (n < NNODES) unsafeAtomicAdd(&counts[batch[n]], 1.0f);
}

__global__ void pool_sum_kernel(const float* __restrict__ h, const int* __restrict__ batch,
                                float* __restrict__ sums) {
    long long idx = (long long)blockIdx.x * blockDim.x + threadIdx.x;
    if (idx >= (long long)NNODES * 16) return;          // 64 feats / 4 per thread
    int n = (int)(idx >> 4);
    int c = (int)(idx & 15) << 2;
    int g = batch[n];
    float4 v = *(const float4*)(h + (long long)n * NOUT + c);
    float* o = sums + g * NOUT + c;
    unsafeAtomicAdd(o + 0, v.x);
    unsafeAtomicAdd(o + 1, v.y);
    unsafeAtomicAdd(o + 2, v.z);
    unsafeAtomicAdd(o + 3, v.w);
}

__global__ void pool_div_kernel(float* __restrict__ out, const float* __restrict__ counts) {
    int i = blockIdx.x * blockDim.x + threadIdx.x;
    if (i < NGRAPH * NOUT) out[i] /= fmaxf(counts[i >> 6], 1.0f);
}

// ---------------------------------------------------------------- launch
static inline int gblk(long long n, int b) { return (int)((n + b - 1) / b); }

extern "C" void kernel_launch(void* const* d_in, const int* in_sizes, int n_in,
                              void* d_out, int out_size, void* d_ws, size_t ws_size,
                              hipStream_t stream) {
    const float* x     = (const float*)d_in[0];
    const int*   src   = (const int*)d_in[1];
    const int*   dst   = (const int*)d_in[2];
    const int*   batch = (const int*)d_in[3];
    const float* W_fc1 = (const float*)d_in[4];
    const float* b_fc1 = (const float*)d_in[5];
    const float* W_c1  = (const float*)d_in[6];
    const float* b_c1  = (const float*)d_in[7];
    const float* W_fc2 = (const float*)d_in[8];
    const float* b_fc2 = (const float*)d_in[9];
    const float* W_c2  = (const float*)d_in[10];
    const float* b_c2  = (const float*)d_in[11];
    const float* W_fc  = (const float*)d_in[12];
    const float* b_fc  = (const float*)d_in[13];
    float* out = (float*)d_out;

    float* bufA   = (float*)d_ws;                          // NNODES*128
    float* bufB   = bufA + (size_t)NNODES * NFEAT;         // NNODES*128
    float* dinv   = bufB + (size_t)NNODES * NFEAT;         // NNODES
    float* counts = dinv + NNODES;                         // 64

    const long long NF = (long long)NNODES * NFEAT;
    const int gemmBlocks = gblk(NNODES, 64);

    // degree / norm
    zero4_kernel<<<gblk(NNODES / 4, 256), 256, 0, stream>>>((float4*)dinv, NNODES / 4);
    deg_kernel<<<gblk(NEDGES, 256), 256, 0, stream>>>(dst, dinv, NEDGES);
    dinv_kernel<<<gblk(NNODES, 256), 256, 0, stream>>>(dinv, NNODES);

    // h = relu(x @ W_fc1 + b_fc1)
    gemm_wmma<128, true, true><<<gemmBlocks, 256, 0, stream>>>(x, W_fc1, b_fc1, bufA, NNODES);
    // conv1: t = h @ W_c1 ; agg = scatter(norm * t[src]) ; h = relu(agg + t*dinv^2 + b_c1)
    gemm_wmma<128, false, false><<<gemmBlocks, 256, 0, stream>>>(bufA, W_c1, nullptr, bufB, NNODES);
    zero4_kernel<<<gblk(NF / 4, 256), 256, 0, stream>>>((float4*)bufA, NF / 4);
    scatter_kernel<<<gblk((long long)NEDGES * 32, 256), 256, 0, stream>>>(bufB, src, dst, dinv, bufA);
    combine_kernel<<<gblk((long long)NNODES * 32, 256), 256, 0, stream>>>(bufA, bufB, dinv, b_c1);

    // h = relu(h @ W_fc2 + b_fc2)
    gemm_wmma<128, true, true><<<gemmBlocks, 256, 0, stream>>>(bufA, W_fc2, b_fc2, bufB, NNODES);
    // conv2
    gemm_wmma<128, false, false><<<gemmBlocks, 256, 0, stream>>>(bufB, W_c2, nullptr, bufA, NNODES);
    zero4_kernel<<<gblk(NF / 4, 256), 256, 0, stream>>>((float4*)bufB, NF / 4);
    scatter_kernel<<<gblk((long long)NEDGES * 32, 256), 256, 0, stream>>>(bufA, src, dst, dinv, bufB);
    combine_kernel<<<gblk((long long)NNODES * 32, 256), 256, 0, stream>>>(bufB, bufA, dinv, b_c2);

    // h = h @ W_fc + b_fc  -> [N, 64]
    gemm_wmma<64, true, false><<<gemmBlocks, 128, 0, stream>>>(bufB, W_fc, b_fc, bufA, NNODES);

    // global mean pool
    zero4_kernel<<<gblk((NGRAPH * NOUT) / 4, 256), 256, 0, stream>>>((float4*)out, (NGRAPH * NOUT) / 4);
    zero4_kernel<<<1, 32, 0, stream>>>((float4*)counts, NGRAPH / 4);
    pool_count_kernel<<<gblk(NNODES, 256), 256, 0, stream>>>(batch, counts);
    pool_sum_kernel<<<gblk((long long)NNODES * 16, 256), 256, 0, stream>>>(bufA, batch, out);
    pool_div_kernel<<<gblk(NGRAPH * NOUT, 256), 256, 0, stream>>>(out, counts);
}